// CPlayer_34626026341131
// MI455X (gfx1250) — compile-verified
//
#include <hip/hip_runtime.h>

#define IN_FEA 256
#define RANK   64
#define HIDDEN 256

typedef __attribute__((ext_vector_type(2))) float v2f;
typedef __attribute__((ext_vector_type(8))) float v8f;

// -------- float atomic multiply via CAS (no native HW op) --------
__device__ __forceinline__ void atomic_mul_f32(float* addr, float v) {
    unsigned int* ua = (unsigned int*)addr;
    unsigned int assumed, old = *ua;
    do {
        assumed = old;
        float newv = __uint_as_float(assumed) * v;
        old = atomicCAS(ua, assumed, __float_as_uint(newv));
    } while (old != assumed);
}

// -------- init: prod = 1.0, deg = 0 --------
__global__ void __launch_bounds__(256) init_kernel(float* __restrict__ prod,
                                                   int* __restrict__ deg,
                                                   int n_nodes) {
    int i = blockIdx.x * blockDim.x + threadIdx.x;
    if (i < n_nodes * RANK) prod[i] = 1.0f;
    if (i < n_nodes) deg[i] = 0;
}

// -------- stage 1: h = x @ W  (f32 WMMA 16x16x4) --------
// One wave computes a 16-row strip of h over all 64 output columns (4 tiles).
// Fragment layouts (ISA 7.12.2, 32-bit):
//   A 16x4: lane l -> row m=l%16, kbase=(l/16)*2; vgpr v = A[m][kk+kbase+v]
//   B 4x16: lane l -> col n=l%16;  vgpr v = B[kk+kbase+v][n]
//   C/D 16x16: lane l -> N=l%16; vgpr v -> M=v+(l/16)*8
// W is staged in LDS PAIR-INTERLEAVED: sW[k/2][c][k%2], so each B fragment is
// one aligned ds_load_b64 straight into the WMMA operand VGPR pair (no movs).
__global__ void __launch_bounds__(128) proj_kernel(const float* __restrict__ x,
                                                   const float* __restrict__ W,
                                                   float* __restrict__ h,
                                                   int n_strips) {
    __shared__ float sW[(IN_FEA / 2) * (2 * RANK)];  // 128*128 f32 = 64 KB

    const int tid = threadIdx.x;
    // repack W[k][c] -> sW[k/2][c*2 + k%2]
    for (int p = tid; p < (IN_FEA / 2) * RANK; p += 128) {
        int c  = p & (RANK - 1);
        int k2 = p >> 6;
        v2f w;
        w.x = W[(2 * k2)     * RANK + c];
        w.y = W[(2 * k2 + 1) * RANK + c];
        *(v2f*)(sW + k2 * (2 * RANK) + c * 2) = w;
    }
    __syncthreads();

    const int wave = tid >> 5;
    const int lane = tid & 31;
    const int strip = blockIdx.x * 4 + wave;
    if (strip >= n_strips) return;  // wave-uniform: EXEC all-1s for WMMA

    const int row0 = strip * 16;
    const int m  = lane & 15;          // A row / B col / C col
    const int kb = (lane >> 4) << 1;   // 0 or 2

    const float* xrow = x + (size_t)(row0 + m) * IN_FEA + kb;  // 8B aligned

    v8f acc0 = {}, acc1 = {}, acc2 = {}, acc3 = {};

    for (int kk = 0; kk < IN_FEA; kk += 4) {
        v2f a = *(const v2f*)(xrow + kk);

        const float* wr = sW + ((kk + kb) >> 1) * (2 * RANK) + m * 2;
        v2f b0 = *(const v2f*)(wr + 0);    // col tile 0: cols  0..15
        v2f b1 = *(const v2f*)(wr + 32);   // col tile 1: cols 16..31
        v2f b2 = *(const v2f*)(wr + 64);   // col tile 2: cols 32..47
        v2f b3 = *(const v2f*)(wr + 96);   // col tile 3: cols 48..63

        acc0 = __builtin_amdgcn_wmma_f32_16x16x4_f32(false, a, false, b0, (short)0, acc0, false, false);
        acc1 = __builtin_amdgcn_wmma_f32_16x16x4_f32(false, a, false, b1, (short)0, acc1, false, false);
        acc2 = __builtin_amdgcn_wmma_f32_16x16x4_f32(false, a, false, b2, (short)0, acc2, false, false);
        acc3 = __builtin_amdgcn_wmma_f32_16x16x4_f32(false, a, false, b3, (short)0, acc3, false, false);
    }

    float* hp = h + (size_t)row0 * RANK;
    const int mb = (lane >> 4) * 8;
#pragma unroll
    for (int v = 0; v < 8; ++v) {
        float* r = hp + (size_t)(mb + v) * RANK + m;
        r[0]  = acc0[v];
        r[16] = acc1[v];
        r[32] = acc2[v];
        r[48] = acc3[v];
    }
}

// -------- stage 2: edge scatter: prod[dst] *= h[src], deg[dst] += 1 --------
__global__ void __launch_bounds__(256) edge_kernel(const int* __restrict__ src,
                                                   const int* __restrict__ dst,
                                                   const float* __restrict__ h,
                                                   float* __restrict__ prod,
                                                   int* __restrict__ deg,
                                                   int n_edges) {
    long long g = (long long)blockIdx.x * blockDim.x + threadIdx.x;
    int e = (int)(g >> 6);   // RANK == 64 features per edge
    int f = (int)(g & 63);
    if (e >= n_edges) return;
    int s = src[e];
    int d = dst[e];
    float val = h[(size_t)s * RANK + f];
    atomic_mul_f32(&prod[(size_t)d * RANK + f], val);
    if (f == 0) atomicAdd(&deg[d], 1);
}

// -------- stage 3: out = (deg>0 ? prod : 0) @ V^T  (f32 WMMA 16x16x4) --------
// B[r][col] = V[col][r] (contiguous in r). The 64-column group slice of V
// (16 KB) is staged in LDS pair-interleaved over r: sV[r/2][c*2 + r%2],
// so each B fragment is a single aligned ds_load_b64.
__global__ void __launch_bounds__(128) out_kernel(const float* __restrict__ prod,
                                                  const int* __restrict__ deg,
                                                  const float* __restrict__ V,
                                                  float* __restrict__ out,
                                                  int n_strips) {
    __shared__ float sV[(RANK / 2) * 128];  // 32*128 f32 = 16 KB

    const int tid = threadIdx.x;
    const int cg = blockIdx.y;  // column group: cols [cg*64, cg*64+64)

    // repack V[cg*64 + c][r] -> sV[r/2][c*2 + r%2]; pairs contiguous in global
    for (int p = tid; p < (RANK / 2) * 64; p += 128) {
        int r2 = p & 31;
        int c  = p >> 5;
        v2f v = *(const v2f*)(V + (size_t)(cg * 64 + c) * RANK + r2 * 2);
        *(v2f*)(sV + r2 * 128 + c * 2) = v;
    }
    __syncthreads();

    const int wave = tid >> 5;
    const int lane = tid & 31;
    const int strip = blockIdx.x * 4 + wave;
    if (strip >= n_strips) return;  // wave-uniform

    const int row0 = strip * 16;
    const int m  = lane & 15;
    const int kb = (lane >> 4) << 1;

    const float mask = (deg[row0 + m] > 0) ? 1.0f : 0.0f;
    const float* arow = prod + (size_t)(row0 + m) * RANK + kb;  // 8B aligned

    v8f acc0 = {}, acc1 = {}, acc2 = {}, acc3 = {};

    for (int kk = 0; kk < RANK; kk += 4) {
        v2f a = *(const v2f*)(arow + kk);
        a = a * mask;

        const float* vr = sV + ((kk + kb) >> 1) * 128 + m * 2;
        v2f b0 = *(const v2f*)(vr + 0);
        v2f b1 = *(const v2f*)(vr + 32);
        v2f b2 = *(const v2f*)(vr + 64);
        v2f b3 = *(const v2f*)(vr + 96);

        acc0 = __builtin_amdgcn_wmma_f32_16x16x4_f32(false, a, false, b0, (short)0, acc0, false, false);
        acc1 = __builtin_amdgcn_wmma_f32_16x16x4_f32(false, a, false, b1, (short)0, acc1, false, false);
        acc2 = __builtin_amdgcn_wmma_f32_16x16x4_f32(false, a, false, b2, (short)0, acc2, false, false);
        acc3 = __builtin_amdgcn_wmma_f32_16x16x4_f32(false, a, false, b3, (short)0, acc3, false, false);
    }

    float* op = out + (size_t)row0 * HIDDEN + cg * 64;
    const int mb = (lane >> 4) * 8;
#pragma unroll
    for (int v = 0; v < 8; ++v) {
        float* r = op + (size_t)(mb + v) * HIDDEN + m;
        r[0]  = acc0[v];
        r[16] = acc1[v];
        r[32] = acc2[v];
        r[48] = acc3[v];
    }
}

extern "C" void kernel_launch(void* const* d_in, const int* in_sizes, int n_in,
                              void* d_out, int out_size, void* d_ws, size_t ws_size,
                              hipStream_t stream) {
    const float* x   = (const float*)d_in[0];
    const float* W   = (const float*)d_in[1];
    const float* V   = (const float*)d_in[2];
    const int*   src = (const int*)d_in[3];
    const int*   dst = (const int*)d_in[4];
    float* out = (float*)d_out;

    const int n_nodes  = in_sizes[0] / IN_FEA;  // 100000
    const int n_edges  = in_sizes[3];           // 1600000
    const int n_strips = n_nodes / 16;          // 6250 (exact)

    // workspace: h [n*64 f32] | prod [n*64 f32] | deg [n i32]  (~51.6 MB)
    char* ws = (char*)d_ws;
    float* h    = (float*)ws;
    float* prod = (float*)(ws + (size_t)n_nodes * RANK * sizeof(float));
    int*   deg  = (int*)(ws + 2 * (size_t)n_nodes * RANK * sizeof(float));

    {   // init prod = 1, deg = 0
        int total  = n_nodes * RANK;
        int blocks = (total + 255) / 256;
        init_kernel<<<blocks, 256, 0, stream>>>(prod, deg, n_nodes);
    }
    {   // h = x @ W
        int blocks = (n_strips + 3) / 4;
        proj_kernel<<<blocks, 128, 0, stream>>>(x, W, h, n_strips);
    }
    {   // segment product + degree
        long long total = (long long)n_edges * RANK;
        int blocks = (int)((total + 255) / 256);
        edge_kernel<<<blocks, 256, 0, stream>>>(src, dst, h, prod, deg, n_edges);
    }
    {   // out = masked(prod) @ V^T
        dim3 grid((n_strips + 3) / 4, HIDDEN / 64);
        out_kernel<<<grid, 128, 0, stream>>>(prod, deg, V, out, n_strips);
    }
}